// Implicit_420906795143
// MI455X (gfx1250) — compile-verified
//
#include <hip/hip_runtime.h>
#include <math.h>

// Problem constants (match reference): z = tanh(z @ W.T + x), B=256, D=384.
#define BROWS 256
#define DDIM  384
#define ITERS 400              // Picard iterations; contraction -> residual << 1e-4
#define PITCH (DDIM + 4)       // LDS row pitch: 388 mod 64 = 4 -> conflict-free A reads

typedef __attribute__((ext_vector_type(2))) float v2f;
typedef __attribute__((ext_vector_type(8))) float v8f;

// ---------------------------------------------------------------------------
// One-time transpose: Wt[k][n] = W[n][k], so WMMA B-fragments load coalesced.
// ---------------------------------------------------------------------------
__global__ __launch_bounds__(256) void wt_transpose(const float* __restrict__ W,
                                                    float* __restrict__ Wt) {
    __shared__ float tile[32][33];
    const int bx = blockIdx.x * 32;   // col block in W (k index)
    const int by = blockIdx.y * 32;   // row block in W (n index)
    const int tx = threadIdx.x;       // 0..31
    const int ty = threadIdx.y;       // 0..7
#pragma unroll
    for (int i = 0; i < 32; i += 8)
        tile[ty + i][tx] = W[(by + ty + i) * DDIM + bx + tx];
    __syncthreads();
#pragma unroll
    for (int i = 0; i < 32; i += 8)
        Wt[(bx + ty + i) * DDIM + by + tx] = tile[tx][ty + i];
}

// ---------------------------------------------------------------------------
// Persistent per-row-tile fixed-point solver.
// Block = 256 threads (8 wave32). Each block owns 16 batch rows in LDS and
// runs ITERS iterations of  z <- tanh(z @ W.T + x)  using v_wmma_f32_16x16x4_f32.
// Each wave produces 3 adjacent 16x16 N-tiles (8 waves * 3 * 16 = 384 cols).
// ---------------------------------------------------------------------------
__global__ __launch_bounds__(256) void implicit_fixed_point(
        const float* __restrict__ x,
        const float* __restrict__ Wt,   // [DDIM][DDIM] transposed W
        float* __restrict__ out) {

    __shared__ float Zs[16][PITCH];
    __shared__ float Xs[16][PITCH];

    const int t  = threadIdx.x;
    const int m0 = blockIdx.x * 16;

    // ---- Load x tile, init z = tanh(x). 256 threads * 24 floats = 16*384. ----
    {
        const int row = t >> 4;            // 0..15
        const int c0  = (t & 15) * 24;     // 24 floats per thread
        const float4* src = reinterpret_cast<const float4*>(x + (m0 + row) * DDIM + c0);
#pragma unroll
        for (int i = 0; i < 6; ++i) {
            float4 v = src[i];
            const int c = c0 + 4 * i;
            Xs[row][c + 0] = v.x; Xs[row][c + 1] = v.y;
            Xs[row][c + 2] = v.z; Xs[row][c + 3] = v.w;
            Zs[row][c + 0] = tanhf(v.x); Zs[row][c + 1] = tanhf(v.y);
            Zs[row][c + 2] = tanhf(v.z); Zs[row][c + 3] = tanhf(v.w);
        }
    }
    __syncthreads();

    const int wave  = t >> 5;          // 0..7
    const int lane  = t & 31;
    const int lrow  = lane & 15;       // M row (A/C) or N col (B/C) within tile
    const int half  = lane >> 4;       // 0: lanes 0-15, 1: lanes 16-31
    const int khalf = half << 1;       // K offset 0 or 2 (A/B fragment layout)
    const int n0base = wave * 48;      // this wave covers cols [n0base, n0base+48)

    // Per-lane base pointer into Wt for B fragments: row (k+khalf), col n.
    const float* wbase = Wt + (size_t)khalf * DDIM + lrow;

#pragma unroll 1
    for (int it = 0; it < ITERS; ++it) {
        v8f acc[3];
#pragma unroll
        for (int j = 0; j < 3; ++j)
#pragma unroll
            for (int r = 0; r < 8; ++r) acc[j][r] = 0.0f;

#pragma unroll 8
        for (int k = 0; k < DDIM; k += 4) {
            v2f a;                              // A: 16x4 tile of z rows
            a.x = Zs[lrow][k + khalf];
            a.y = Zs[lrow][k + khalf + 1];
            const float* wp = wbase + (size_t)k * DDIM;
#pragma unroll
            for (int j = 0; j < 3; ++j) {
                const int n0 = n0base + j * 16;
                v2f b;                          // B: 4x16 tile of W^T (coalesced rows of Wt)
                b.x = wp[n0];
                b.y = wp[n0 + DDIM];
                acc[j] = __builtin_amdgcn_wmma_f32_16x16x4_f32(
                    /*neg_a=*/false, a, /*neg_b=*/false, b,
                    /*c_mod=*/(short)0, acc[j],
                    /*reuse_a=*/false, /*reuse_b=*/false);
            }
        }

        // ---- epilogue: znew = tanh(acc + x); double barrier around Zs update ----
        float nv[3][8];
#pragma unroll
        for (int j = 0; j < 3; ++j) {
            const int n = n0base + j * 16 + lrow;
#pragma unroll
            for (int r = 0; r < 8; ++r) {
                const int m = r + (half << 3);        // C/D layout: VGPR r -> rows r, r+8
                nv[j][r] = tanhf(acc[j][r] + Xs[m][n]);
            }
        }
        __syncthreads();   // all waves done reading Zs
#pragma unroll
        for (int j = 0; j < 3; ++j) {
            const int n = n0base + j * 16 + lrow;
#pragma unroll
            for (int r = 0; r < 8; ++r)
                Zs[r + (half << 3)][n] = nv[j][r];
        }
        __syncthreads();   // Zs fully updated before next iteration reads
    }

    // ---- Final z tile IS the reference output tanh(z* W^T + x). Store it. ----
    {
        const int row = t >> 4;
        const int c0  = (t & 15) * 24;
        float4* dst = reinterpret_cast<float4*>(out + (m0 + row) * DDIM + c0);
#pragma unroll
        for (int i = 0; i < 6; ++i) {
            const int c = c0 + 4 * i;
            float4 v;
            v.x = Zs[row][c + 0]; v.y = Zs[row][c + 1];
            v.z = Zs[row][c + 2]; v.w = Zs[row][c + 3];
            dst[i] = v;
        }
    }
}

// ---------------------------------------------------------------------------
extern "C" void kernel_launch(void* const* d_in, const int* in_sizes, int n_in,
                              void* d_out, int out_size, void* d_ws, size_t ws_size,
                              hipStream_t stream) {
    const float* x = (const float*)d_in[0];   // [256, 384] f32
    const float* W = (const float*)d_in[1];   // [384, 384] f32
    float* out = (float*)d_out;               // [256, 384] f32
    float* Wt  = (float*)d_ws;                // [384, 384] f32 scratch (576 KB)

    dim3 tb(32, 8);
    dim3 tg(DDIM / 32, DDIM / 32);
    wt_transpose<<<tg, tb, 0, stream>>>(W, Wt);

    implicit_fixed_point<<<BROWS / 16, 256, 0, stream>>>(x, Wt, out);
}